// temporal_attention_layer_33225867001900
// MI455X (gfx1250) — compile-verified
//
#include <hip/hip_runtime.h>
#include <math.h>

// CDNA5 WMMA fragment types
typedef __attribute__((ext_vector_type(2))) float v2f;
typedef __attribute__((ext_vector_type(8))) float v8f;

#define BATCH    2048
#define LDIM     512
#define JDIM     256
#define NTHREADS 256
#define INV_SQRT_L 0.04419417382415922f  // 1/sqrt(512)

__device__ __forceinline__ float sigmoidf_fast(float v) {
    return 1.0f / (1.0f + __expf(-v));
}

// One workgroup per batch row b. 256 threads = 8 wave32.
// Pass 1: tmp1[j] = sigmoid(sum_l w1[l]*x[b,l,j] + b1[j])      (x from HBM, b128 loads)
// Pass 2: tmp2[l] = sigmoid(tmp1 . w2[:,l] + b2[l])            (WMMA f32 16x16x4, 4-tile ILP)
//         attn    = softmax(tmp2 / sqrt(L))                    (LDS tree reductions)
// Pass 3: h[j]    = sum_l attn[l]*x[b,l,j]                     (x re-read, L2-resident)
__global__ __launch_bounds__(NTHREADS)
void fused_temporal_attention(const float* __restrict__ x,
                              const float* __restrict__ w1,
                              const float* __restrict__ b1,
                              const float* __restrict__ w2,
                              const float* __restrict__ b2,
                              float* __restrict__ h_out,
                              float* __restrict__ attn_out)
{
    __shared__ float  w1_s[LDIM];                       // 2 KB; recycled as reduction scratch
    __shared__ __align__(16) float tmp1_s[JDIM];        // 1 KB (read as v2f in pass 2)
    __shared__ float  tmp2_s[LDIM];                     // 2 KB; tmp2 -> attn in place
    __shared__ float4 part_s[NTHREADS];                 // 4 KB partial-sum combine

    const int b   = blockIdx.x;
    const int tid = threadIdx.x;
    const float* __restrict__ xb = x + (size_t)b * (size_t)(LDIM * JDIM);

    // Preload w1 into LDS (broadcast-read in pass 1 inner loop)
    for (int l = tid; l < LDIM; l += NTHREADS) w1_s[l] = w1[l];
    __syncthreads();

    const int lsub = tid >> 6;          // 0..3 : which L-phase this thread covers
    const int j4   = (tid & 63) << 2;   // 0,4,...,252 : float4 column group

    // ------------------------- pass 1: tmp1 -------------------------
    {
        float ax = 0.f, ay = 0.f, az = 0.f, aw = 0.f;
        #pragma unroll 4
        for (int l = lsub; l < LDIM; l += 4) {
            const float4 xv = *(const float4*)(xb + l * JDIM + j4);
            const float  w  = w1_s[l];
            ax = fmaf(w, xv.x, ax);
            ay = fmaf(w, xv.y, ay);
            az = fmaf(w, xv.z, az);
            aw = fmaf(w, xv.w, aw);
        }
        part_s[tid] = make_float4(ax, ay, az, aw);
    }
    __syncthreads();
    if (tid < 64) {
        const float4 s0 = part_s[tid];
        const float4 s1 = part_s[tid + 64];
        const float4 s2 = part_s[tid + 128];
        const float4 s3 = part_s[tid + 192];
        const int j = tid << 2;
        tmp1_s[j + 0] = sigmoidf_fast(s0.x + s1.x + s2.x + s3.x + b1[j + 0]);
        tmp1_s[j + 1] = sigmoidf_fast(s0.y + s1.y + s2.y + s3.y + b1[j + 1]);
        tmp1_s[j + 2] = sigmoidf_fast(s0.z + s1.z + s2.z + s3.z + b1[j + 2]);
        tmp1_s[j + 3] = sigmoidf_fast(s0.w + s1.w + s2.w + s3.w + b1[j + 3]);
    }
    __syncthreads();

    // ------------------- pass 2: tmp2 via WMMA f32 16x16x4 -------------------
    // A (16x4 f32): lane m = lane&15, kh = lane>>4; VGPR0 = A[m][k+2kh], VGPR1 = A[m][k+2kh+1].
    // Row 0 of A holds tmp1; rows 1..15 zeroed with a mask-multiply (no exec divergence).
    // B (4x16 f32): same K striping, N = lane&15. Each wave owns cols [wave*64, wave*64+64)
    // as 4 independent tiles -> 4 accumulators, 1 shared A load + 8 B loads + 4 wmma per k.
    {
        const int   lane  = tid & 31;
        const int   wave  = tid >> 5;           // 0..7
        const int   m     = lane & 15;
        const int   kh    = lane >> 4;          // 0 or 1
        const float amask = (m == 0) ? 1.0f : 0.0f;
        const int   nbase = wave * 64 + m;
        v8f c0 = {}, c1 = {}, c2 = {}, c3 = {};
        #pragma unroll 4
        for (int k = 0; k < JDIM; k += 4) {
            const int k0 = k + 2 * kh;          // even -> 8B-aligned LDS v2f load
            v2f a = *(const v2f*)(tmp1_s + k0);
            a *= amask;
            const float* __restrict__ w2r = w2 + k0 * LDIM + nbase;
            v2f bf0, bf1, bf2, bf3;
            bf0[0] = w2r[0];   bf0[1] = w2r[LDIM];
            bf1[0] = w2r[16];  bf1[1] = w2r[LDIM + 16];
            bf2[0] = w2r[32];  bf2[1] = w2r[LDIM + 32];
            bf3[0] = w2r[48];  bf3[1] = w2r[LDIM + 48];
            c0 = __builtin_amdgcn_wmma_f32_16x16x4_f32(false, a, false, bf0, (short)0, c0, false, false);
            c1 = __builtin_amdgcn_wmma_f32_16x16x4_f32(false, a, false, bf1, (short)0, c1, false, false);
            c2 = __builtin_amdgcn_wmma_f32_16x16x4_f32(false, a, false, bf2, (short)0, c2, false, false);
            c3 = __builtin_amdgcn_wmma_f32_16x16x4_f32(false, a, false, bf3, (short)0, c3, false, false);
        }
        if (lane < 16) {  // D row 0: M=0 lives in c[0] on lanes 0..15 (N = lane)
            const int n = wave * 64 + lane;
            tmp2_s[n]      = sigmoidf_fast(c0[0] + b2[n])      * INV_SQRT_L;
            tmp2_s[n + 16] = sigmoidf_fast(c1[0] + b2[n + 16]) * INV_SQRT_L;
            tmp2_s[n + 32] = sigmoidf_fast(c2[0] + b2[n + 32]) * INV_SQRT_L;
            tmp2_s[n + 48] = sigmoidf_fast(c3[0] + b2[n + 48]) * INV_SQRT_L;
        }
    }
    __syncthreads();

    // ------------------------- softmax over L -------------------------
    // max
    w1_s[tid] = fmaxf(tmp2_s[tid], tmp2_s[tid + 256]);
    __syncthreads();
    for (int off = 128; off > 0; off >>= 1) {
        if (tid < off) w1_s[tid] = fmaxf(w1_s[tid], w1_s[tid + off]);
        __syncthreads();
    }
    const float rowmax = w1_s[0];
    __syncthreads();
    // exp + sum
    const float e0 = __expf(tmp2_s[tid] - rowmax);
    const float e1 = __expf(tmp2_s[tid + 256] - rowmax);
    w1_s[tid] = e0 + e1;
    __syncthreads();
    for (int off = 128; off > 0; off >>= 1) {
        if (tid < off) w1_s[tid] += w1_s[tid + off];
        __syncthreads();
    }
    const float inv = 1.0f / w1_s[0];
    const float a0 = e0 * inv;
    const float a1 = e1 * inv;
    tmp2_s[tid]        = a0;
    tmp2_s[tid + 256]  = a1;
    attn_out[(size_t)b * LDIM + tid]       = a0;
    attn_out[(size_t)b * LDIM + tid + 256] = a1;
    __syncthreads();

    // ------------------------- pass 3: h -------------------------
    {
        float ax = 0.f, ay = 0.f, az = 0.f, aw = 0.f;
        #pragma unroll 4
        for (int l = lsub; l < LDIM; l += 4) {
            const float4 xv = *(const float4*)(xb + l * JDIM + j4);
            const float  w  = tmp2_s[l];
            ax = fmaf(w, xv.x, ax);
            ay = fmaf(w, xv.y, ay);
            az = fmaf(w, xv.z, az);
            aw = fmaf(w, xv.w, aw);
        }
        part_s[tid] = make_float4(ax, ay, az, aw);
    }
    __syncthreads();
    if (tid < 64) {
        const float4 s0 = part_s[tid];
        const float4 s1 = part_s[tid + 64];
        const float4 s2 = part_s[tid + 128];
        const float4 s3 = part_s[tid + 192];
        const int j = tid << 2;
        float* __restrict__ hb = h_out + (size_t)b * JDIM;
        hb[j + 0] = s0.x + s1.x + s2.x + s3.x;
        hb[j + 1] = s0.y + s1.y + s2.y + s3.y;
        hb[j + 2] = s0.z + s1.z + s2.z + s3.z;
        hb[j + 3] = s0.w + s1.w + s2.w + s3.w;
    }
}

extern "C" void kernel_launch(void* const* d_in, const int* in_sizes, int n_in,
                              void* d_out, int out_size, void* d_ws, size_t ws_size,
                              hipStream_t stream) {
    (void)in_sizes; (void)n_in; (void)out_size; (void)d_ws; (void)ws_size;
    const float* x  = (const float*)d_in[0];  // (B, L, J)
    const float* w1 = (const float*)d_in[1];  // (1, L)
    const float* b1 = (const float*)d_in[2];  // (1, J)
    const float* w2 = (const float*)d_in[3];  // (J, L)
    const float* b2 = (const float*)d_in[4];  // (1, L)

    float* h_out    = (float*)d_out;                          // (B, J)
    float* attn_out = (float*)d_out + (size_t)BATCH * JDIM;   // (B, L)

    fused_temporal_attention<<<dim3(BATCH), dim3(NTHREADS), 0, stream>>>(
        x, w1, b1, w2, b2, h_out, attn_out);
}